// CalculateSLayer_76536317215120
// MI455X (gfx1250) — compile-verified
//
#include <hip/hip_runtime.h>

// Problem constants (reference: N=4096 tokens, D_S=70 features, 2 adj channels)
#define NN   4096
#define DD   70
#define NT   256        // 16-wide tiles along the node dimension
#define ND   5          // 16-wide tiles along the feature dimension (70 -> 80 padded)
#define SPLITK 8        // split the 4096-long reduction across 8 workgroups
#define TPC  (NT / SPLITK)

typedef float v2f __attribute__((ext_vector_type(2)));
typedef float v4f __attribute__((ext_vector_type(4)));
typedef float v8f __attribute__((ext_vector_type(8)));

// D = A(16x4, f32) * B(4x16, f32) + C(16x16, f32)  -- full-precision CDNA5 WMMA
__device__ __forceinline__ v8f wmma4(v2f a, v2f b, v8f c) {
  return __builtin_amdgcn_wmma_f32_16x16x4_f32(
      /*neg_a=*/false, a, /*neg_b=*/false, b,
      /*c_mod=*/(short)0, c, /*reuse_a=*/false, /*reuse_b=*/false);
}

__global__ void zero_f32(float* __restrict__ p, int n) {
  int i = blockIdx.x * blockDim.x + threadIdx.x;
  if (i < n) p[i] = 0.0f;
}

// Repack s (4096x70 row-major f32) into exact WMMA-B operand lane order:
//   sB[((t*ND + n)*4 + q)*32 + lane] = { s[r, d], s[r+1, d] }
//   r = 16t + 4q + 2*(lane>>4), d = 16n + (lane&15); zero-pad d >= 70.
// The hot loop then loads B with one coalesced 8-byte load per lane.
__global__ void pack_sB(const float* __restrict__ s, v2f* __restrict__ sB) {
  int tid = blockIdx.x * blockDim.x + threadIdx.x;
  if (tid >= NT * ND * 4 * 32) return;
  int lane = tid & 31;
  int e = tid >> 5;
  int q = e & 3;  e >>= 2;
  int n = e % ND;
  int t = e / ND;
  int r = 16 * t + 4 * q + 2 * (lane >> 4);
  int d = 16 * n + (lane & 15);
  v2f v;
  v.x = (d < DD) ? s[(size_t)r * DD + d]       : 0.0f;
  v.y = (d < DD) ? s[(size_t)(r + 1) * DD + d] : 0.0f;
  sB[tid] = v;
}

// One wave32 per workgroup. Block b owns output rows [16b,16b+16) of BOTH
// s_in (= a^T s) and s_out (= a s); chunk c sweeps t in [32c, 32c+32).
// adj[i,j,k]: a[i,j] = adj[(i*NN+j)*2] + adj[(i*NN+j)*2+1] (channel sum fused
// into the A-operand build).
__global__ __launch_bounds__(32) void slayer_wmma(const float* __restrict__ adj,
                                                  const v2f*  __restrict__ sB,
                                                  float*      __restrict__ out) {
  const int wg   = blockIdx.x;
  const int b    = wg >> 3;         // output row-block 0..255
  const int c    = wg & (SPLITK-1); // reduction chunk 0..7
  const int lane = threadIdx.x;
  const int m    = lane & 15;       // M (A) / N (B) index for this lane
  const int kh   = lane >> 4;       // K-half select

  v8f zero = {};
  v8f accI[ND], accO[ND];
#pragma unroll
  for (int n = 0; n < ND; ++n) { accI[n] = zero; accO[n] = zero; }

  const int i0 = 16 * b;
  for (int t = TPC * c; t < TPC * c + TPC; ++t) {
    const int j0 = 16 * t;
#pragma unroll
    for (int q = 0; q < 4; ++q) {
      const int kb = 4 * q + 2 * kh;
      // A operand for s_out: a[i0+m, j0+kb .. +1]; b128 grabs both channels of
      // two adjacent j -> channel-sum in VALU. Lanes l and l+16 read adjacent
      // 16B -> 32B contiguous per row.
      v4f f = *(const v4f*)(adj + 2 * ((size_t)(i0 + m) * NN + (j0 + kb)));
      v2f aO; aO.x = f.x + f.y; aO.y = f.z + f.w;
      // A operand for s_in (a^T tile): rows j0+kb, j0+kb+1, column i0+m.
      // 16 lanes read 16 consecutive j-columns -> 128B contiguous per row.
      const float* p = adj + 2 * ((size_t)(j0 + kb) * NN + (i0 + m));
      v2f g0 = *(const v2f*)p;
      v2f g1 = *(const v2f*)(p + 2 * NN);
      v2f aI; aI.x = g0.x + g0.y; aI.y = g1.x + g1.y;
      // Shared B operand: s[j0.., dtile], pre-packed in lane order.
      const v2f* bp = sB + (size_t)t * (ND * 4 * 32) + q * 32 + lane;
#pragma unroll
      for (int n = 0; n < ND; ++n) {
        v2f bv = bp[n * 4 * 32];
        accO[n] = wmma4(aO, bv, accO[n]);  // s_out += a * s
        accI[n] = wmma4(aI, bv, accI[n]);  // s_in  += a^T * s
      }
    }
  }

  // Epilogue: accumulate split-K partials into d_out (zeroed by zero_f32).
  // C/D layout: vgpr r, lane l -> M = r + 8*(l>>4), N = l&15.
  float* outI = out;                       // s_in  : first NN*DD floats
  float* outO = out + (size_t)NN * DD;     // s_out : next  NN*DD floats
#pragma unroll
  for (int n = 0; n < ND; ++n) {
    int d = 16 * n + m;
    if (d < DD) {
#pragma unroll
      for (int r = 0; r < 8; ++r) {
        int row = i0 + r + 8 * kh;
        atomicAdd(&outI[(size_t)row * DD + d], accI[n][r]);
        atomicAdd(&outO[(size_t)row * DD + d], accO[n][r]);
      }
    }
  }
}

extern "C" void kernel_launch(void* const* d_in, const int* in_sizes, int n_in,
                              void* d_out, int out_size, void* d_ws, size_t ws_size,
                              hipStream_t stream) {
  const float* adj = (const float*)d_in[0];   // (4096, 4096, 2) f32
  const float* s   = (const float*)d_in[1];   // (4096, 70) f32
  float* out = (float*)d_out;                 // [s_in | s_out], 2*4096*70 f32
  v2f* sB = (v2f*)d_ws;                       // 256*5*4*32 float2 = 1.31 MB

  zero_f32<<<(out_size + 255) / 256, 256, 0, stream>>>(out, out_size);

  int packN = NT * ND * 4 * 32;
  pack_sB<<<(packN + 255) / 256, 256, 0, stream>>>(s, sB);

  slayer_wmma<<<NT * SPLITK, 32, 0, stream>>>(adj, sB, out);
}